// Net_58729382805608
// MI455X (gfx1250) — compile-verified
//
#include <hip/hip_runtime.h>

#define N_F_IN   512
#define N_HIDDEN 64
#define N_CLS    47
#define K_HOPS_N 10
#define ALPHA_C  0.1f

typedef __attribute__((ext_vector_type(2))) float v2f;
typedef __attribute__((ext_vector_type(8))) float v8f;

// ---------------------------------------------------------------------------
// h0 = relu(x @ W_embed + b_embed)   [N,512] @ [512,64]
// block = 128 threads = 4 waves; block owns 16 rows, each wave one 16-col tile.
// x tile staged in LDS (stride 65 -> conflict-free A-fragment reads).
// ---------------------------------------------------------------------------
__global__ void __launch_bounds__(128) k_embed(const float* __restrict__ x,
                                               const float* __restrict__ W,
                                               const float* __restrict__ bias,
                                               float* __restrict__ h0, int N) {
  __shared__ float lds[16 * 65];
  const int row0 = blockIdx.x * 16;
  const int wave = threadIdx.x >> 5;
  const int lane = threadIdx.x & 31;
  const int half = lane >> 4;   // 0: K,K+1   1: K+2,K+3
  const int l16  = lane & 15;
  const int col0 = wave * 16;

  v8f acc = {};
  for (int kc = 0; kc < N_F_IN; kc += 64) {
    __syncthreads();
    #pragma unroll
    for (int i = threadIdx.x; i < 16 * 64; i += 128) {
      const int r = i >> 6, c = i & 63;
      const int rg = row0 + r;
      lds[r * 65 + c] = (rg < N) ? x[(size_t)rg * N_F_IN + kc + c] : 0.f;
    }
    __syncthreads();
    #pragma unroll
    for (int k = 0; k < 64; k += 4) {
      const int ka = k + half * 2;
      v2f a, b;
      // A 16x4: lanes 0-15 rows M, K=ka..ka+1; lanes 16-31 K=ka+2..ka+3
      a.x = lds[l16 * 65 + ka];
      a.y = lds[l16 * 65 + ka + 1];
      // B 4x16: VGPR0 = rows K (lanes 0-15) / K+2 (lanes 16-31), VGPR1 = K+1/K+3
      b.x = W[(size_t)(kc + ka) * N_HIDDEN + col0 + l16];
      b.y = W[(size_t)(kc + ka + 1) * N_HIDDEN + col0 + l16];
      acc = __builtin_amdgcn_wmma_f32_16x16x4_f32(false, a, false, b,
                                                  (short)0, acc, false, false);
    }
  }

  const int colg = col0 + l16;
  const float bv = bias[colg];
  #pragma unroll
  for (int r = 0; r < 8; ++r) {
    const int rg = row0 + half * 8 + r;   // C/D: lanes<16 rows 0-7, lanes>=16 rows 8-15
    if (rg < N) {
      float v = acc[r] + bv;
      h0[(size_t)rg * N_HIDDEN + colg] = v > 0.f ? v : 0.f;
    }
  }
}

// ---------------------------------------------------------------------------
// degree / normalization
// ---------------------------------------------------------------------------
__global__ void k_zero(float* __restrict__ p, int n) {
  const int i = blockIdx.x * blockDim.x + threadIdx.x;
  if (i < n) p[i] = 0.f;
}

__global__ void k_deg(const int* __restrict__ col, float* __restrict__ deg, int E) {
  const int i = blockIdx.x * blockDim.x + threadIdx.x;
  if (i < E) unsafeAtomicAdd(&deg[col[i]], 1.0f);
}

// in-place deg -> dis = rsqrt(deg + 1)   (+1 = self loop; always >= 1)
__global__ void k_dis(float* __restrict__ d, int n) {
  const int i = blockIdx.x * blockDim.x + threadIdx.x;
  if (i < n) d[i] = rsqrtf(d[i] + 1.0f);
}

// ---------------------------------------------------------------------------
// one hop:  h_next = ALPHA*h0 + (1-ALPHA)*[ dis_i^2 * h_i  +  sum_edges ]
// init handles the self-loop + teleport terms densely (no atomics needed)
// ---------------------------------------------------------------------------
__global__ void k_hop_init(const float* __restrict__ h0, const float* __restrict__ h,
                           const float* __restrict__ dis, float* __restrict__ hn,
                           int n64) {
  const int i = blockIdx.x * blockDim.x + threadIdx.x;
  if (i < n64) {
    const float d = dis[i >> 6];
    hn[i] = ALPHA_C * h0[i] + (1.0f - ALPHA_C) * d * d * h[i];
  }
}

// one wave per edge; lane handles a float2 of the 64-wide feature row
__global__ void k_hop_scatter(const int* __restrict__ row, const int* __restrict__ col,
                              const float* __restrict__ dis, const float* __restrict__ h,
                              float* __restrict__ hn, int E) {
  const int wid  = (blockIdx.x * blockDim.x + threadIdx.x) >> 5;
  const int lane = threadIdx.x & 31;
  if (wid >= E) return;
  const int r = row[wid];
  const int c = col[wid];
  const float w = (1.0f - ALPHA_C) * dis[r] * dis[c];
  const float2 v = ((const float2*)(h + (size_t)r * N_HIDDEN))[lane];
  float* out = hn + (size_t)c * N_HIDDEN + lane * 2;
  unsafeAtomicAdd(out,     w * v.x);   // global_atomic_add_f32
  unsafeAtomicAdd(out + 1, w * v.y);
}

// ---------------------------------------------------------------------------
// logits = h @ W_pred + b_pred   [N,64] @ [64,47]; one wave per 16x16 tile
// ---------------------------------------------------------------------------
__global__ void __launch_bounds__(32) k_pred(const float* __restrict__ h,
                                             const float* __restrict__ W,
                                             const float* __restrict__ bias,
                                             float* __restrict__ logits, int N) {
  const int row0 = blockIdx.x * 16;
  const int col0 = blockIdx.y * 16;
  const int lane = threadIdx.x & 31;
  const int half = lane >> 4;
  const int l16  = lane & 15;
  const int colg = col0 + l16;
  const bool cv  = colg < N_CLS;
  const int ra   = row0 + l16;
  const bool rv  = ra < N;

  v8f acc = {};
  #pragma unroll
  for (int k = 0; k < N_HIDDEN; k += 4) {
    const int ka = k + half * 2;
    v2f a, b;
    a.x = rv ? h[(size_t)ra * N_HIDDEN + ka]     : 0.f;
    a.y = rv ? h[(size_t)ra * N_HIDDEN + ka + 1] : 0.f;
    b.x = cv ? W[(size_t)ka * N_CLS + colg]       : 0.f;
    b.y = cv ? W[(size_t)(ka + 1) * N_CLS + colg] : 0.f;
    acc = __builtin_amdgcn_wmma_f32_16x16x4_f32(false, a, false, b,
                                                (short)0, acc, false, false);
  }
  if (!cv) return;
  const float bv = bias[colg];
  #pragma unroll
  for (int r = 0; r < 8; ++r) {
    const int rg = row0 + half * 8 + r;
    if (rg < N) logits[(size_t)rg * N_CLS + colg] = acc[r] + bv;
  }
}

// ---------------------------------------------------------------------------
// softmax / log_softmax: one wave per node (47 classes; lane covers c, c+32)
// ---------------------------------------------------------------------------
__global__ void __launch_bounds__(256) k_softmax(const float* __restrict__ logits,
                                                 float* __restrict__ out_log,
                                                 float* __restrict__ out_soft, int N) {
  const int wid  = (blockIdx.x * blockDim.x + threadIdx.x) >> 5;
  const int lane = threadIdx.x & 31;
  if (wid >= N) return;
  const size_t base = (size_t)wid * N_CLS;
  const float v0 = logits[base + lane];
  const bool  h1 = (lane + 32) < N_CLS;
  const float v1 = h1 ? logits[base + lane + 32] : -3.0e38f;

  float m = fmaxf(v0, v1);
  #pragma unroll
  for (int off = 16; off > 0; off >>= 1)
    m = fmaxf(m, __shfl_xor(m, off, 32));

  const float e0 = expf(v0 - m);
  const float e1 = h1 ? expf(v1 - m) : 0.f;
  float s = e0 + e1;
  #pragma unroll
  for (int off = 16; off > 0; off >>= 1)
    s += __shfl_xor(s, off, 32);

  const float ls   = logf(s);
  const float rinv = 1.0f / s;
  out_log [base + lane] = v0 - m - ls;
  out_soft[base + lane] = e0 * rinv;
  if (h1) {
    out_log [base + lane + 32] = v1 - m - ls;
    out_soft[base + lane + 32] = e1 * rinv;
  }
}

// ---------------------------------------------------------------------------
extern "C" void kernel_launch(void* const* d_in, const int* in_sizes, int n_in,
                              void* d_out, int out_size, void* d_ws, size_t ws_size,
                              hipStream_t stream) {
  const float* x  = (const float*)d_in[0];
  const int*   ei = (const int*)  d_in[1];
  const float* We = (const float*)d_in[2];
  const float* be = (const float*)d_in[3];
  const float* Wp = (const float*)d_in[4];
  const float* bp = (const float*)d_in[5];

  const int N = in_sizes[0] / N_F_IN;
  const int E = in_sizes[1] / 2;
  const int* row = ei;       // source
  const int* col = ei + E;   // target

  const size_t n64 = (size_t)N * N_HIDDEN;
  float* h0  = (float*)d_ws;
  float* hA  = h0 + n64;
  float* hB  = hA + n64;
  float* dis = hB + n64;

  float* out_log  = (float*)d_out;
  float* logits   = out_log + (size_t)N * N_CLS;
  float* out_soft = logits  + (size_t)N * N_CLS;

  // 1) embed + relu (WMMA f32)
  k_embed<<<(N + 15) / 16, 128, 0, stream>>>(x, We, be, h0, N);

  // 2) gcn norm
  k_zero<<<(N + 255) / 256, 256, 0, stream>>>(dis, N);
  k_deg <<<(E + 255) / 256, 256, 0, stream>>>(col, dis, E);
  k_dis <<<(N + 255) / 256, 256, 0, stream>>>(dis, N);

  // 3) K_HOPS propagation (ping-pong buffers; deterministic sequence)
  const float* hcur = h0;
  float* bufs[2] = {hA, hB};
  const int ni = (int)n64;
  const long long tw = (long long)E * 32;           // one wave per edge
  const int scatter_blocks = (int)((tw + 255) / 256);
  for (int hop = 0; hop < K_HOPS_N; ++hop) {
    float* hnext = bufs[hop & 1];
    k_hop_init<<<(ni + 255) / 256, 256, 0, stream>>>(h0, hcur, dis, hnext, ni);
    k_hop_scatter<<<scatter_blocks, 256, 0, stream>>>(row, col, dis, hcur, hnext, E);
    hcur = hnext;
  }

  // 4) prediction head (WMMA f32), 3 col tiles cover 47 classes
  dim3 pg((N + 15) / 16, 3);
  k_pred<<<pg, 32, 0, stream>>>(hcur, Wp, bp, logits, N);

  // 5) softmax / log_softmax
  k_softmax<<<(N + 7) / 8, 256, 0, stream>>>(logits, out_log, out_soft, N);
}